// LowPass_66090956751487
// MI455X (gfx1250) — compile-verified
//
#include <hip/hip_runtime.h>

// ---------------------------------------------------------------------------
// filtfilt (order-2 Butterworth, odd extension, per-row max normalization)
// for x of shape (320, T).  Exact block-parallel IIR:
//   - chunk decomposition (L samples) + 2x2 state scan across chunks
//   - within a chunk, 16x16 tiles: Y = Gamma * F + boundary, Gamma constant
//     lower-triangular impulse-response matrix -> 4 chained
//     V_WMMA_F32_16X16X4_F32 per tile (CDNA5 wave32 WMMA).
//   - tile loop split into slow (edge-mapped) prologue/epilogue and a
//     branch-free fast interior path.
// ---------------------------------------------------------------------------

typedef __attribute__((ext_vector_type(2))) float v2f;
typedef __attribute__((ext_vector_type(8))) float v8f;

#define ROWS 320
#define PAD  9          // padlen = 3*max(len(a),len(b)) = 9

__device__ __forceinline__ float lane_shfl(float v, int src) {
  return __shfl(v, src, 32);
}

// Slow-path fetch of u[t] in the (odd-extended, normalized) signal domain.
// DIR==0: forward input = odd_ext(x/scale).  DIR==1: reversed y1.
template<int DIR>
__device__ __forceinline__ float fetchU(const float* __restrict__ x,
                                        const float* __restrict__ y1,
                                        const float* __restrict__ edges,
                                        int row, int T, float inv, int t) {
  const int Te = T + 2 * PAD;
  if (t < 0 || t >= Te) return 0.f;
  if (DIR == 0) {
    const float* xr = x + (size_t)row * T;
    float v;
    if (t < PAD)           v = 2.f * xr[0]     - xr[PAD - t];
    else if (t < PAD + T)  v = xr[t - PAD];
    else                   v = 2.f * xr[T - 1] - xr[T - 2 - (t - PAD - T)];
    return v * inv;
  } else {
    int j = Te - 1 - t;                 // forward-time index into y1 (padded)
    if (j < PAD)       return edges[row * 2 * PAD + j];
    if (j < PAD + T)   return y1[(size_t)row * T + (j - PAD)];
    return edges[row * 2 * PAD + PAD + (j - PAD - T)];
  }
}

// Fast-path fetch: whole window known to map to the row interior.
// rowp = x + row*T (DIR0) or y1 + row*T (DIR1); base = Te-1-PAD.
template<int DIR>
__device__ __forceinline__ float fastU(const float* __restrict__ rowp,
                                       int base, float inv, int t) {
  return (DIR == 0) ? rowp[t - PAD] * inv : rowp[base - t];
}

// Slow-path store.
template<int DIR>
__device__ __forceinline__ void storeY(float* __restrict__ out,
                                       float* __restrict__ edges,
                                       int row, int T, float scale,
                                       int t, float val) {
  const int Te = T + 2 * PAD;
  if (t < 0 || t >= Te) return;
  if (DIR == 0) {
    if (t < PAD)          edges[row * 2 * PAD + t] = val;
    else if (t < PAD + T) out[(size_t)row * T + (t - PAD)] = val;
    else                  edges[row * 2 * PAD + PAD + (t - PAD - T)] = val;
  } else {
    int tau = Te - 1 - t;
    if (tau >= PAD && tau < PAD + T)
      out[(size_t)row * T + (tau - PAD)] = val * scale;
  }
}

// --------------------------- per-row max ------------------------------------
__global__ __launch_bounds__(256) void rowMaxKernel(const float* __restrict__ x,
                                                    float* __restrict__ sc,
                                                    float* __restrict__ inv,
                                                    int T) {
  const int row = blockIdx.x;
  const float* xr = x + (size_t)row * T;
  float m = 0.f;
  if ((T & 3) == 0) {
    const float4* x4 = reinterpret_cast<const float4*>(xr);
    for (int i = threadIdx.x; i < (T >> 2); i += blockDim.x) {
      float4 w = x4[i];
      m = fmaxf(m, fmaxf(fmaxf(fabsf(w.x), fabsf(w.y)),
                         fmaxf(fabsf(w.z), fabsf(w.w))));
    }
  } else {
    for (int i = threadIdx.x; i < T; i += blockDim.x)
      m = fmaxf(m, fabsf(xr[i]));
  }
  __shared__ float red[256];
  red[threadIdx.x] = m;
  __syncthreads();
  for (int s = 128; s > 0; s >>= 1) {
    if (threadIdx.x < s) red[threadIdx.x] = fmaxf(red[threadIdx.x], red[threadIdx.x + s]);
    __syncthreads();
  }
  if (threadIdx.x == 0) { sc[row] = red[0]; inv[row] = 1.f / red[0]; }
}

// ----------------- cross-chunk 2x2 linear state scan ------------------------
// state s = (y[n], y[n-1]); A = [[-a1,-a2],[1,0]]; M = A^L (L power of two).
__global__ __launch_bounds__(ROWS) void scanKernel(const float* __restrict__ aArr,
                                                   const float* __restrict__ vbuf,
                                                   float* __restrict__ sbuf,
                                                   int C, int L) {
  const int row = threadIdx.x;
  if (row >= ROWS) return;
  const double ia0 = 1.0 / (double)aArr[0];
  const double a1 = (double)aArr[1] * ia0, a2 = (double)aArr[2] * ia0;
  double m00 = -a1, m01 = -a2, m10 = 1.0, m11 = 0.0;
  for (int p = L; p > 1; p >>= 1) {          // M = A^(2^log2(L))
    double n00 = m00 * m00 + m01 * m10, n01 = m00 * m01 + m01 * m11;
    double n10 = m10 * m00 + m11 * m10, n11 = m10 * m01 + m11 * m11;
    m00 = n00; m01 = n01; m10 = n10; m11 = n11;
  }
  const float M00 = (float)m00, M01 = (float)m01, M10 = (float)m10, M11 = (float)m11;
  float s1 = 0.f, s2 = 0.f;                  // state BEFORE chunk 0
  for (int c = 0; c < C; ++c) {
    size_t idx = ((size_t)c * ROWS + row) * 2;
    sbuf[idx] = s1; sbuf[idx + 1] = s2;
    float v1 = vbuf[idx], v2 = vbuf[idx + 1];
    float n1 = M00 * s1 + M01 * s2 + v1;
    float n2 = M10 * s1 + M11 * s2 + v2;
    s1 = n1; s2 = n2;
  }
}

// --------------------------- one 16x16 tile ---------------------------------
template<int DIR, int MODE, bool FAST>
__device__ __forceinline__ void doTile(
    const float* __restrict__ x, float* __restrict__ out,
    float* __restrict__ edges, const float* __restrict__ rowp, int base,
    int row, int T, int Te, float inv, float scl, int ts, int h, int n,
    const v2f (&Aslab)[4], const float (&pb1)[8], const float (&pb2)[8],
    float b0, float b1, float b2,
    float& c1, float& c2, float& yp1, float& yp2) {
  // next-tile FIR input carries, fetched BEFORE this tile's in-place stores
  float nc2, nc1;
  if (FAST) {
    nc2 = fastU<DIR>(rowp, base, inv, ts + 14);
    nc1 = fastU<DIR>(rowp, base, inv, ts + 15);
  } else {
    nc2 = fetchU<DIR>(x, out, edges, row, T, inv, ts + 14);
    nc1 = fetchU<DIR>(x, out, edges, row, T, inv, ts + 15);
  }

  // B matrix: F[k][row] = b0*u[ts+k] + b1*u[ts+k-1] + b2*u[ts+k-2],
  // lane half h covers K slots {4s+2h, 4s+2h+1}.
  v2f Bs[4];
#pragma unroll
  for (int s = 0; s < 4; ++s) {
    int tb = ts + 4 * s + 2 * h - 2;
    float w0, w1, w2, w3;
    if (FAST) {
      w0 = fastU<DIR>(rowp, base, inv, tb);
      w1 = fastU<DIR>(rowp, base, inv, tb + 1);
      w2 = fastU<DIR>(rowp, base, inv, tb + 2);
      w3 = fastU<DIR>(rowp, base, inv, tb + 3);
    } else {
      w0 = fetchU<DIR>(x, out, edges, row, T, inv, tb);
      w1 = fetchU<DIR>(x, out, edges, row, T, inv, tb + 1);
      w2 = fetchU<DIR>(x, out, edges, row, T, inv, tb + 2);
      w3 = fetchU<DIR>(x, out, edges, row, T, inv, tb + 3);
    }
    if (s == 0) {            // lanes h==0 (K=0,1) need u[ts-2], u[ts-1] carries
      w0 = (h == 0) ? c2 : w0;
      w1 = (h == 0) ? c1 : w1;
    }
    Bs[s].x = b0 * w2 + b1 * w1 + b2 * w0;
    Bs[s].y = b0 * w3 + b1 * w2 + b2 * w1;
  }

  // C = boundary contribution of incoming output state
  v8f acc;
#pragma unroll
  for (int v = 0; v < 8; ++v) acc[v] = pb1[v] * yp1 + pb2[v] * yp2;

  // D = Gamma * F + C : 4 chained V_WMMA_F32_16X16X4_F32
  acc = __builtin_amdgcn_wmma_f32_16x16x4_f32(false, Aslab[0], false, Bs[0],
                                              (short)0, acc, false, false);
  acc = __builtin_amdgcn_wmma_f32_16x16x4_f32(false, Aslab[1], false, Bs[1],
                                              (short)0, acc, false, false);
  acc = __builtin_amdgcn_wmma_f32_16x16x4_f32(false, Aslab[2], false, Bs[2],
                                              (short)0, acc, false, false);
  acc = __builtin_amdgcn_wmma_f32_16x16x4_f32(false, Aslab[3], false, Bs[3],
                                              (short)0, acc, false, false);

  // outgoing state = y[ts+15], y[ts+14] (regs 7,6 of upper-half lanes)
  yp2 = lane_shfl(acc[6], 16 + n);
  yp1 = lane_shfl(acc[7], 16 + n);

  if (MODE == 1) {
#pragma unroll
    for (int v = 0; v < 8; ++v) {
      int t = ts + v + 8 * h;
      if (FAST) {
        if (DIR == 0) out[(size_t)row * T + (t - PAD)] = acc[v];
        else          out[(size_t)row * T + (base - t)] = acc[v] * scl;
      } else {
        storeY<DIR>(out, edges, row, T, scl, t, acc[v]);
      }
    }
  }
  c1 = nc1; c2 = nc2;
}

// ------------------------- WMMA IIR chunk pass ------------------------------
// One wave (32 lanes) per (16-row tile, chunk).  MODE 0: compute zero-state
// final state v per chunk (+ cache chunk-boundary inputs in bbuf).
// MODE 1: recompute with exact initial state from sbuf and store outputs.
template<int DIR, int MODE>
__global__ __launch_bounds__(32) void iirPass(
    const float* __restrict__ x, const float* __restrict__ bArr,
    const float* __restrict__ aArr, float* __restrict__ out,
    const float* __restrict__ scArr, const float* __restrict__ invArr,
    float* __restrict__ edges, float* __restrict__ vbuf,
    const float* __restrict__ sbuf, float* __restrict__ bbuf,
    int T, int L) {
  const int lane = threadIdx.x;
  const int n = lane & 15;      // M index (time-in-tile) for A; row for B/D
  const int h = lane >> 4;      // lane half selects K slots {2h, 2h+1} per slab
  const int row = blockIdx.y * 16 + n;
  const int c = blockIdx.x;
  const int Te = T + 2 * PAD;
  const int cStart = c * L;
  const int len = (Te - cStart < L) ? (Te - cStart) : L;
  const int nTiles = (len + 15) >> 4;
  const int base = Te - 1 - PAD;
  const float* rowp = (DIR == 0 ? x : (const float*)out) + (size_t)row * T;

  const float ia0 = 1.f / aArr[0];
  const float a1 = aArr[1] * ia0, a2 = aArr[2] * ia0;
  const float b0 = bArr[0] * ia0, b1 = bArr[1] * ia0, b2 = bArr[2] * ia0;
  const float inv = invArr[row];
  const float scl = scArr[row];

  // impulse response h[j] of 1/A(z): lane (l&15) keeps h[l&15]
  float myh;
  {
    float hj = 1.f, hjm1 = 0.f;
    myh = (n == 0) ? 1.f : 0.f;
#pragma unroll
    for (int j = 1; j < 16; ++j) {
      float t = -a1 * hj - a2 * hjm1; hjm1 = hj; hj = t;
      if (j == n) myh = hj;
    }
  }

  // A-matrix (Gamma, lower-triangular Toeplitz) in WMMA f32 16x16x4 layout:
  // lanes 0-15 M=0..15 with K={4s,4s+1}, lanes 16-31 M=0..15 with K={4s+2,4s+3}.
  v2f Aslab[4];
#pragma unroll
  for (int s = 0; s < 4; ++s) {
    int k0 = 4 * s + 2 * h;
    int i0 = n - k0, i1 = n - k0 - 1;
    float g0 = lane_shfl(myh, i0 < 0 ? 0 : i0);
    float g1 = lane_shfl(myh, i1 < 0 ? 0 : i1);
    Aslab[s].x = (i0 >= 0) ? g0 : 0.f;
    Aslab[s].y = (i1 >= 0) ? g1 : 0.f;
  }

  // boundary-state propagation coefficients per accumulator register
  // (C/D reg v holds time n = v + 8h):  p1=-a1*h[m]-a2*h[m-1], p2=-a2*h[m]
  float pb1[8], pb2[8];
#pragma unroll
  for (int v = 0; v < 8; ++v) {
    int m = v + 8 * h;
    float hm  = lane_shfl(myh, m);
    float hm1 = lane_shfl(myh, m >= 1 ? m - 1 : 0);
    if (m < 1) hm1 = 0.f;
    pb1[v] = -a1 * hm - a2 * hm1;
    pb2[v] = -a2 * hm;
  }

  const size_t sidx = ((size_t)c * ROWS + row) * 2;

  float yp1, yp2;   // y[ts-1], y[ts-2]  (output state)
  float c1, c2;     // u[ts-1], u[ts-2]  (FIR input carry, used by h==0 lanes)
  if (MODE == 0) {
    yp1 = 0.f; yp2 = 0.f;
    c1 = fetchU<DIR>(x, out, edges, row, T, inv, cStart - 1);
    c2 = fetchU<DIR>(x, out, edges, row, T, inv, cStart - 2);
    if (h == 0) { bbuf[sidx] = c1; bbuf[sidx + 1] = c2; }
  } else {
    yp1 = sbuf[sidx]; yp2 = sbuf[sidx + 1];
    c1 = bbuf[sidx];  c2 = bbuf[sidx + 1];
  }

  // tile ranges: fast iff whole window [ts-2, ts+15] maps to the row interior
  //   <=>  ts >= PAD+2  &&  ts + 16 + PAD <= Te      (same for both DIRs)
  int tFastLo = (cStart >= PAD + 2) ? 0 : ((PAD + 2 - cStart + 15) >> 4);
  if (tFastLo > nTiles) tFastLo = nTiles;
  int hi = Te - 16 - PAD - cStart;                 // max fast (ts - cStart)
  int tFastHi = (hi < 0) ? 0 : ((hi >> 4) + 1);
  if (tFastHi > nTiles) tFastHi = nTiles;
  if (tFastHi < tFastLo) tFastHi = tFastLo;

  int tile = 0, ts = cStart;
  for (; tile < tFastLo; ++tile, ts += 16)
    doTile<DIR, MODE, false>(x, out, edges, rowp, base, row, T, Te, inv, scl,
                             ts, h, n, Aslab, pb1, pb2, b0, b1, b2,
                             c1, c2, yp1, yp2);
  for (; tile < tFastHi; ++tile, ts += 16) {
    if (tile + 3 < tFastHi)   // stays strictly inside the fast region
      __builtin_prefetch(rowp + (DIR == 0 ? (ts + 48 - PAD) : (base - (ts + 63))), 0, 1);
    doTile<DIR, MODE, true>(x, out, edges, rowp, base, row, T, Te, inv, scl,
                            ts, h, n, Aslab, pb1, pb2, b0, b1, b2,
                            c1, c2, yp1, yp2);
  }
  for (; tile < nTiles; ++tile, ts += 16)
    doTile<DIR, MODE, false>(x, out, edges, rowp, base, row, T, Te, inv, scl,
                             ts, h, n, Aslab, pb1, pb2, b0, b1, b2,
                             c1, c2, yp1, yp2);

  if (MODE == 0 && h == 0) {           // zero-state final chunk state
    vbuf[sidx] = yp1; vbuf[sidx + 1] = yp2;
  }
}

// ---------------------------------------------------------------------------
extern "C" void kernel_launch(void* const* d_in, const int* in_sizes, int n_in,
                              void* d_out, int out_size, void* d_ws, size_t ws_size,
                              hipStream_t stream) {
  const float* x = (const float*)d_in[0];
  const float* b = (const float*)d_in[1];
  const float* a = (const float*)d_in[2];
  float* out = (float*)d_out;
  const int T  = in_sizes[0] / ROWS;
  const int Te = T + 2 * PAD;

  // pick power-of-two chunk length so workspace fits (base: L=1024)
  int L = 1024, C;
  const size_t fixedF = 2 * ROWS + (size_t)ROWS * 2 * PAD;
  for (;;) {
    C = (Te + L - 1) / L;
    size_t totF = fixedF + 3ull * ROWS * C * 2;
    if (totF * sizeof(float) <= ws_size || L >= (1 << 20)) break;
    L <<= 1;
  }

  float* ws    = (float*)d_ws;
  float* sc    = ws;                               // ROWS
  float* inv   = sc + ROWS;                        // ROWS
  float* edges = inv + ROWS;                       // ROWS*18 (y1 pad samples)
  float* vbuf  = edges + (size_t)ROWS * 2 * PAD;   // ROWS*C*2 zero-state states
  float* sbuf  = vbuf + (size_t)ROWS * C * 2;      // ROWS*C*2 scanned states
  float* bbuf  = sbuf + (size_t)ROWS * C * 2;      // ROWS*C*2 boundary inputs

  rowMaxKernel<<<ROWS, 256, 0, stream>>>(x, sc, inv, T);

  dim3 grid(C, ROWS / 16), blk(32);
  // forward filter: x -> y1 (interior in d_out, pads in `edges`)
  iirPass<0, 0><<<grid, blk, 0, stream>>>(x, b, a, out, sc, inv, edges, vbuf, sbuf, bbuf, T, L);
  scanKernel<<<1, ROWS, 0, stream>>>(a, vbuf, sbuf, C, L);
  iirPass<0, 1><<<grid, blk, 0, stream>>>(x, b, a, out, sc, inv, edges, vbuf, sbuf, bbuf, T, L);
  // backward filter over reversed y1, rescale, keep interior (in place)
  iirPass<1, 0><<<grid, blk, 0, stream>>>(x, b, a, out, sc, inv, edges, vbuf, sbuf, bbuf, T, L);
  scanKernel<<<1, ROWS, 0, stream>>>(a, vbuf, sbuf, C, L);
  iirPass<1, 1><<<grid, blk, 0, stream>>>(x, b, a, out, sc, inv, edges, vbuf, sbuf, bbuf, T, L);
}